// HungarianMatcher_11261404250566
// MI455X (gfx1250) — compile-verified
//
#include <hip/hip_runtime.h>
#include <hip/hip_bf16.h>
#include <math.h>

// Problem constants (match reference)
constexpr int Bb = 8;
constexpr int Nn = 300;
constexpr int Cc = 92;         // num classes
constexpr int Tt = 30;         // targets
constexpr int PADSTRIDE = 304; // padded square cost row stride (>=300, 16B aligned)
constexpr int MT = 19;         // ceil(300/16) M tiles
constexpr int NT = 2;          // ceil(30/16) N tiles

typedef __attribute__((ext_vector_type(2))) float v2f;
typedef __attribute__((ext_vector_type(4))) float v4f;
typedef __attribute__((ext_vector_type(8))) float v8f;

// ---------------------------------------------------------------------------
// Kernel 0: softmax over the class dim. One wave (32 lanes) per row of 92.
// ---------------------------------------------------------------------------
__global__ void softmax92_kernel(const float* __restrict__ logits,
                                 float* __restrict__ probs) {
  const int row = blockIdx.x * blockDim.y + threadIdx.y;   // 0 .. B*N-1
  if (row >= Bb * Nn) return;
  const int lane = threadIdx.x;                             // 0..31 (wave32)
  const float* in = logits + (size_t)row * Cc;

  float mx = -INFINITY;
  for (int k = lane; k < Cc; k += 32) mx = fmaxf(mx, in[k]);
  #pragma unroll
  for (int off = 16; off; off >>= 1) mx = fmaxf(mx, __shfl_xor(mx, off, 32));

  float e[3];
  int cnt = 0;
  float s = 0.f;
  for (int k = lane; k < Cc; k += 32) { float t = expf(in[k] - mx); e[cnt++] = t; s += t; }
  #pragma unroll
  for (int off = 16; off; off >>= 1) s += __shfl_xor(s, off, 32);

  const float inv = 1.f / s;
  cnt = 0;
  for (int k = lane; k < Cc; k += 32) probs[(size_t)row * Cc + k] = e[cnt++] * inv;
}

// ---------------------------------------------------------------------------
// Kernel 1: cost matrix via WMMA f32 16x16x4 (one-hot class GEMM, exact in f32)
// fused with L1 bbox cost + GIoU cost in the epilogue.
// Grid = B*MT*NT blocks of 32 threads (one wave per 16x16 output tile).
// A-fragment rows are CLAMPED (not predicated): out-of-range rows only feed
// D elements that the epilogue never writes, so no exec-masked loads needed —
// each K-step's A pair is one unconditional 8B-aligned global_load_b64.
// ---------------------------------------------------------------------------
__global__ void cost_wmma_kernel(const float* __restrict__ probs,
                                 const float* __restrict__ pred_boxes,
                                 const int*   __restrict__ tgt_labels,
                                 const float* __restrict__ tgt_boxes,
                                 float* __restrict__ outC,    // [B,N,T] -> d_out[0..71999]
                                 float* __restrict__ Cpad) {  // [B,N,PADSTRIDE]
  const int lane = threadIdx.x;        // 0..31
  int tile = blockIdx.x;
  const int nt = tile % NT;  tile /= NT;
  const int mt = tile % MT;
  const int b  = tile / MT;
  const int m0 = mt * 16, n0 = nt * 16;

  const int half = lane >> 4;          // 0 or 1
  const int l16  = lane & 15;

  // A fragment source: probs row m (16x4 f32: lane -> M=l16, K pair = half*2 + r)
  const int mrow  = m0 + l16;
  const int mclmp = (mrow < Nn) ? mrow : (Nn - 1);          // safe addr; result unused
  const float* prow = probs + ((size_t)b * Nn + mclmp) * Cc;

  // B fragment source: one-hot(tgt_label) column (4x16 f32: lane -> N=l16, K = half*2 + r)
  const int tcol = n0 + l16;
  const int lbl  = (tcol < Tt) ? tgt_labels[b * Tt + tcol] : -1;

  v8f acc = {};
  #pragma unroll
  for (int k0 = 0; k0 < Cc; k0 += 4) {
    const int kb = k0 + half * 2;
    const v2f a = *(const v2f*)(prow + kb);                 // one global_load_b64
    v2f bb;
    bb[0] = (lbl == kb)     ? 1.f : 0.f;
    bb[1] = (lbl == kb + 1) ? 1.f : 0.f;
    // D = A(16x4, f32) * B(4x16, one-hot f32) + C : exact gather of probs
    acc = __builtin_amdgcn_wmma_f32_16x16x4_f32(
        false, a, false, bb, (short)0, acc, false, false);
  }

  // Target box for this lane's column (cxcywh -> xyxy), one b128 load
  v4f tb = {};
  if (tcol < Tt) tb = *(const v4f*)(tgt_boxes + ((size_t)b * Tt + tcol) * 4);
  const float tcx = tb[0], tcy = tb[1], tw = tb[2], th = tb[3];
  const float tx0 = tcx - 0.5f * tw, ty0 = tcy - 0.5f * th;
  const float tx1 = tcx + 0.5f * tw, ty1 = tcy + 0.5f * th;
  const float ta  = fmaxf(tx1 - tx0, 0.f) * fmaxf(ty1 - ty0, 0.f);

  // Epilogue: D layout is lane -> N = l16, VGPR v -> M = m0 + v + half*8
  #pragma unroll
  for (int v = 0; v < 8; ++v) {
    const int m = m0 + v + half * 8;
    if (m < Nn && tcol < Tt) {
      const v4f pb = *(const v4f*)(pred_boxes + ((size_t)b * Nn + m) * 4); // b128
      const float pcx = pb[0], pcy = pb[1], pw = pb[2], ph = pb[3];
      const float l1 = fabsf(pcx - tcx) + fabsf(pcy - tcy) +
                       fabsf(pw - tw)   + fabsf(ph - th);
      const float px0 = pcx - 0.5f * pw, py0 = pcy - 0.5f * ph;
      const float px1 = pcx + 0.5f * pw, py1 = pcy + 0.5f * ph;
      const float pa  = fmaxf(px1 - px0, 0.f) * fmaxf(py1 - py0, 0.f);
      const float ix0 = fmaxf(px0, tx0), iy0 = fmaxf(py0, ty0);
      const float ix1 = fminf(px1, tx1), iy1 = fminf(py1, ty1);
      const float inter = fmaxf(ix1 - ix0, 0.f) * fmaxf(iy1 - iy0, 0.f);
      const float uni = pa + ta - inter;
      const float iou = inter / fmaxf(uni, 1e-6f);
      const float cx0 = fminf(px0, tx0), cy0 = fminf(py0, ty0);
      const float cx1 = fmaxf(px1, tx1), cy1 = fmaxf(py1, ty1);
      const float ca  = fmaxf(cx1 - cx0, 0.f) * fmaxf(cy1 - cy0, 0.f);
      const float giou = iou - (ca - uni) / fmaxf(ca, 1e-6f);
      // COST_CLASS*(-prob) + COST_BBOX*l1 + COST_GIOU*(-giou)
      const float cost = -acc[v] + 5.f * l1 - 2.f * giou;
      outC[((size_t)b * Nn + m) * Tt + tcol] = cost;
      Cpad[((size_t)b * Nn + m) * PADSTRIDE + tcol] = cost;
    }
  }
}

// ---------------------------------------------------------------------------
// Kernel 2: per-batch max(C)+1 and fill padded columns [T..PADSTRIDE).
// One 256-thread block per batch item.
// ---------------------------------------------------------------------------
__global__ void padfill_kernel(const float* __restrict__ outC,
                               float* __restrict__ Cpad) {
  const int b = blockIdx.x;
  const int tid = threadIdx.x;
  __shared__ float red[256];

  const float* Cb = outC + (size_t)b * Nn * Tt;
  float mx = -INFINITY;
  for (int i = tid; i < Nn * Tt; i += 256) mx = fmaxf(mx, Cb[i]);
  red[tid] = mx;
  __syncthreads();
  for (int s = 128; s > 0; s >>= 1) {
    if (tid < s) red[tid] = fmaxf(red[tid], red[tid + s]);
    __syncthreads();
  }
  const float padv = red[0] + 1.f;

  float* Pb = Cpad + (size_t)b * Nn * PADSTRIDE;
  constexpr int PADW = PADSTRIDE - Tt;             // 274 cols per row
  for (int idx = tid; idx < Nn * PADW; idx += 256) {
    const int i = idx / PADW;
    const int j = Tt + idx % PADW;
    Pb[(size_t)i * PADSTRIDE + j] = padv;
  }
}

// ---------------------------------------------------------------------------
// Kernel 3: Jonker-Volgenant Hungarian, one wave32 per batch item.
// Columns j in [0,300] strided across lanes: j = k*32 + lane, k<10.
// v, minv, used live in registers; u, p, way live in LDS.
// Argmin each step via 5-step shfl_xor (val,idx) reduction.
// ---------------------------------------------------------------------------
__global__ void hungarian_kernel(const float* __restrict__ Cpad,
                                 float* __restrict__ dout) {
  const int b = blockIdx.x;
  const int lane = threadIdx.x;        // 32 threads, one wave
  constexpr int n = Nn;                // 300
  constexpr float INF = 1e18f;

  __shared__ float u[PADSTRIDE];
  __shared__ int   p[PADSTRIDE];
  __shared__ int   way[PADSTRIDE];

  for (int k = lane; k < PADSTRIDE; k += 32) { u[k] = 0.f; p[k] = 0; way[k] = 0; }
  float vloc[10];
  #pragma unroll
  for (int k = 0; k < 10; ++k) vloc[k] = 0.f;
  __syncthreads();

  const float* Cb = Cpad + (size_t)b * Nn * PADSTRIDE;

  for (int i = 1; i <= n; ++i) {
    if (lane == 0) p[0] = i;
    float minv[10];
    unsigned usedmask = 0u;
    #pragma unroll
    for (int k = 0; k < 10; ++k) minv[k] = INF;
    __syncthreads();

    int j0 = 0;
    while (true) {
      const int i0 = p[j0];                    // uniform LDS read
      if (i0 == 0) break;
      if ((j0 & 31) == lane) usedmask |= 1u << (j0 >> 5);  // used[j0] = true
      const float ui0 = u[i0];
      const float* row = Cb + (size_t)(i0 - 1) * PADSTRIDE;

      float lmin = INF;
      int   lidx = 0x7fffffff;
      #pragma unroll
      for (int k = 0; k < 10; ++k) {
        const int j = k * 32 + lane;
        const bool valid = (j >= 1) && (j <= n) && !((usedmask >> k) & 1u);
        if (valid) {
          const float cur = row[j - 1] - ui0 - vloc[k];
          if (cur < minv[k]) { minv[k] = cur; way[j] = j0; }
          if (minv[k] < lmin) { lmin = minv[k]; lidx = j; }  // k asc => first min
        }
      }
      // wave-wide (min, argmin) reduction; ties -> smaller j (argmin semantics)
      #pragma unroll
      for (int off = 16; off; off >>= 1) {
        const float om = __shfl_xor(lmin, off, 32);
        const int   oi = __shfl_xor(lidx, off, 32);
        if (om < lmin || (om == lmin && oi < lidx)) { lmin = om; lidx = oi; }
      }
      const float delta = lmin;
      const int   j1    = lidx;

      #pragma unroll
      for (int k = 0; k < 10; ++k) {
        const int j = k * 32 + lane;
        if ((usedmask >> k) & 1u) {
          const int pj = p[j];     // matched rows are distinct -> no LDS conflict
          u[pj] += delta;
          vloc[k] -= delta;
        } else {
          minv[k] -= delta;        // includes invalid j; harmless
        }
      }
      __syncthreads();
      j0 = j1;
    }

    // augmenting-path reconstruction (serial, lane 0)
    if (lane == 0) {
      while (j0 != 0) {
        const int jp = way[j0];
        p[j0] = p[jp];
        j0 = jp;
      }
    }
    __syncthreads();
  }

  // Outputs: row_ind = p[1..T]-1 at [72000 + b*T], col_ind = 0..T-1 at [72240 + b*T]
  if (lane < Tt) {
    dout[(size_t)Bb * Nn * Tt + b * Tt + lane]            = (float)(p[lane + 1] - 1);
    dout[(size_t)Bb * Nn * Tt + Bb * Tt + b * Tt + lane]  = (float)lane;
  }
}

// ---------------------------------------------------------------------------
// Launch: softmax -> WMMA cost -> pad fill -> Hungarian (all on `stream`)
// ---------------------------------------------------------------------------
extern "C" void kernel_launch(void* const* d_in, const int* in_sizes, int n_in,
                              void* d_out, int out_size, void* d_ws, size_t ws_size,
                              hipStream_t stream) {
  (void)in_sizes; (void)n_in; (void)out_size; (void)ws_size;
  const float* logits  = (const float*)d_in[0];   // [B,N,C]
  const float* pboxes  = (const float*)d_in[1];   // [B,N,4]
  const int*   tlabels = (const int*)  d_in[2];   // [B,T]
  const float* tboxes  = (const float*)d_in[3];   // [B,T,4]
  float* out = (float*)d_out;

  float* probs = (float*)d_ws;                                          // B*N*C f32
  float* Cpad  = (float*)((char*)d_ws + (size_t)Bb * Nn * Cc * sizeof(float)); // B*N*PADSTRIDE f32

  softmax92_kernel<<<dim3((Bb * Nn + 7) / 8), dim3(32, 8), 0, stream>>>(logits, probs);
  cost_wmma_kernel<<<dim3(Bb * MT * NT), dim3(32), 0, stream>>>(
      probs, pboxes, tlabels, tboxes, out, Cpad);
  padfill_kernel<<<dim3(Bb), dim3(256), 0, stream>>>(out, Cpad);
  hungarian_kernel<<<dim3(Bb), dim3(32), 0, stream>>>(Cpad, out);
}